// CriticREM_62182536511994
// MI455X (gfx1250) — compile-verified
//
#include <hip/hip_runtime.h>
#include <hip/hip_bf16.h>

// ---------------------------------------------------------------------------
// CriticREM on MI455X (gfx1250), FP32 WMMA path (V_WMMA_F32_16X16X4_F32).
// Memory-bound problem (34 MB in / 12.9 GFLOP -> ~1.5us HBM floor at 23.3TB/s),
// so we keep full fp32 precision (matches reference numerics) and focus on:
//   (1) WMMA for all GEMM work,
//   (2) killing redundant weight traffic: fragment-major packed W1^T staged in
//       LDS via the Tensor Data Mover (tensor_load_to_lds, TENSORcnt),
//       packed W2^T streamed from L2 with coalesced b64 reads + prefetch,
//   (3) 2 M-tiles per wave so every B-fragment is reused twice from registers.
// ---------------------------------------------------------------------------

typedef __attribute__((ext_vector_type(2))) float v2f;
typedef __attribute__((ext_vector_type(4))) float v4f;
typedef __attribute__((ext_vector_type(8))) float v8f;
typedef __attribute__((ext_vector_type(4))) unsigned int u32x4;
typedef __attribute__((ext_vector_type(4))) int i32x4;
typedef __attribute__((ext_vector_type(8))) int i32x8;

#define BATCH      65536
#define STATE_DIM  96
#define ACTION_DIM 32
#define IN_F       128
#define HID        256
#define NHEADS     200

// fragment-packed weight sizes (dwords)
#define PW1_DW (16 * 32 * 64)   // 16 ntiles * 32 ksteps * (32 lanes * 2) = 32768
#define PW2_DW (16 * 64 * 64)   // 16 ntiles * 64 ksteps * 64            = 65536

// LDS layout (dwords)
#define LDS_W1_DW  PW1_DW       // 128 KB packed W1^T
#define H1_STRIDE  260          // 256 + pad; stride mod 64 banks == 4 -> conflict-free b64
#define H1_PER_WAVE (32 * H1_STRIDE)
#define SMEM_DW    (LDS_W1_DW + 4 * H1_PER_WAVE)   // 66048 dw = 258 KB

// ---------------------------------------------------------------------------
// Kernel 1: combine the 200 heads:  Wc[d] = sum_h a[h]*Wh[h,0,d], bc = sum a*bh
// ---------------------------------------------------------------------------
__global__ __launch_bounds__(256) void critic_prep_head(
    const float* __restrict__ alphas, const float* __restrict__ Wh,
    const float* __restrict__ bh, float* __restrict__ wc, float* __restrict__ bc)
{
    int d = threadIdx.x;            // 0..255
    float acc = 0.f;
    for (int h = 0; h < NHEADS; ++h)
        acc = fmaf(alphas[h], Wh[h * HID + d], acc);
    wc[d] = acc;
    if (d == 0) {
        float s = 0.f;
        for (int h = 0; h < NHEADS; ++h)
            s = fmaf(alphas[h], bh[h], s);
        *bc = s;
    }
}

// ---------------------------------------------------------------------------
// Kernel 2: pack W1^T and W2^T into WMMA f32 B-fragment-major layout.
// Fragment (ntile, kstep): 32 lanes * 2 dwords (= 256 B, one coalesced b64/lane).
// Lane L (<16):  B[k0+0..1][ntile*16+L]     Lane L (>=16): B[k0+2..3][ntile*16+L-16]
// where B[k][n] = W[n][k]  (B = W^T).
// ---------------------------------------------------------------------------
__global__ __launch_bounds__(256) void critic_prep_pack(
    const float* __restrict__ W1, const float* __restrict__ W2,
    float* __restrict__ pw1, float* __restrict__ pw2)
{
    int t = blockIdx.x * 256 + threadIdx.x;
    if (t < PW1_DW) {
        int j    = t & 1;
        int lane = (t >> 1) & 31;
        int ks   = (t >> 6) & 31;
        int nt   = t >> 11;
        int n = nt * 16 + (lane & 15);
        int k = ks * 4 + ((lane >> 4) << 1) + j;
        pw1[t] = W1[n * IN_F + k];
    } else if (t < PW1_DW + PW2_DW) {
        int i = t - PW1_DW;
        int j    = i & 1;
        int lane = (i >> 1) & 31;
        int ks   = (i >> 6) & 63;
        int nt   = i >> 12;
        int n = nt * 16 + (lane & 15);
        int k = ks * 4 + ((lane >> 4) << 1) + j;
        pw2[i] = W2[n * HID + k];
    }
}

// ---------------------------------------------------------------------------
// Kernel 3: main fused MLP. 4 waves/block, each wave owns 2 M-tiles (32 rows),
// block covers 128 rows; 512 blocks cover B=65536 exactly.
// ---------------------------------------------------------------------------
__global__ __launch_bounds__(128, 1) void critic_main(
    const float* __restrict__ state, const float* __restrict__ action,
    const float* __restrict__ b1, const float* __restrict__ b2,
    const float* __restrict__ pw1g, const float* __restrict__ pw2,
    const float* __restrict__ wc, const float* __restrict__ bcp,
    float* __restrict__ out)
{
    extern __shared__ float smem[];
    float* lds_w1 = smem;
    float* lds_h1 = smem + LDS_W1_DW;

    // --- stage packed W1 into LDS via the Tensor Data Mover (one wave issues
    //     a 1-D D# descriptor: 32768 x f32 contiguous copy, DMA -> LDS) ---
#if __has_builtin(__builtin_amdgcn_tensor_load_to_lds)
    if (threadIdx.x < 32) {
        unsigned long long ga = (unsigned long long)pw1g;
        unsigned ldsb = __builtin_amdgcn_groupstaticsize();   // dyn-LDS base (0)
        u32x4 g0 = {
            1u,                                               // count=1 (valid D#)
            ldsb,                                             // lds_addr (bytes)
            (unsigned)(ga & 0xffffffffu),                     // global_addr[31:0]
            (unsigned)((ga >> 32) & 0x1ffffffu) | (2u << 30)  // addr[56:32] | type=2
        };
        i32x8 g1 = {
            (int)(2u << 16),          // wg_mask=0, data_size=2 (4 B), no flags
            (int)(32768u << 16),      // abar_addr=0 | tensor_dim0[15:0]=0x8000
            (int)(1u << 16),          // tensor_dim0[31:16]=0 | tensor_dim1[15:0]=1
            (int)(32768u << 16),      // tensor_dim1[31:16]=0 | tile_dim0=0x8000
            1,                        // tile_dim1=1, tile_dim2=0
            32768,                    // tensor_dim0_stride[31:0]
            0, 0                      // stride hi, tensor_dim1_stride
        };
        i32x4 gz = { 0, 0, 0, 0 };
#if __clang_major__ >= 23
        i32x8 gz8 = { 0, 0, 0, 0, 0, 0, 0, 0 };
        __builtin_amdgcn_tensor_load_to_lds(g0, g1, gz, gz, gz8, 0);
#else
        __builtin_amdgcn_tensor_load_to_lds(g0, g1, gz, gz, 0);
#endif
        __builtin_amdgcn_s_wait_tensorcnt(0);
    }
#else
    for (int i = threadIdx.x; i < LDS_W1_DW / 4; i += 128)
        ((v4f*)lds_w1)[i] = ((const v4f*)pw1g)[i];
#endif
    __syncthreads();

    const int wave = threadIdx.x >> 5;
    const int lane = threadIdx.x & 31;
    const int mloc = lane & 15;
    const int koff = (lane >> 4) << 1;          // A/B fragment K sub-offset
    const long rowbase = (long)blockIdx.x * 128 + wave * 32;

    const float* st0 = state  + (rowbase + mloc) * STATE_DIM + koff;
    const float* st1 = st0 + 16 * STATE_DIM;
    const float* ac0 = action + (rowbase + mloc) * ACTION_DIM + koff;
    const float* ac1 = ac0 + 16 * ACTION_DIM;
    float* h1 = lds_h1 + wave * H1_PER_WAVE;    // [32 rows][260] per-wave tile

    const int roff = (lane < 16) ? 0 : 8;

    // =========================== Layer 1 ===================================
    // h1[32x256] = relu(x[32x128] @ W1^T + b1), x = concat(state, action).
    // K loop split at the concat boundary (k=96 == ks 24) -> no per-lane selects.
    for (int nt = 0; nt < 16; ++nt) {
        v8f c0 = {};
        v8f c1 = {};
        const float* wb = lds_w1 + (nt * 32) * 64 + lane * 2;
#pragma unroll 4
        for (int ks = 0; ks < 24; ++ks) {       // k in [0,96): state
            v2f b  = *(const v2f*)(wb + ks * 64);        // ds_load_b64, no conflicts
            v2f a0 = *(const v2f*)(st0 + ks * 4);
            v2f a1 = *(const v2f*)(st1 + ks * 4);
            c0 = __builtin_amdgcn_wmma_f32_16x16x4_f32(false, a0, false, b, (short)0, c0, false, false);
            c1 = __builtin_amdgcn_wmma_f32_16x16x4_f32(false, a1, false, b, (short)0, c1, false, false);
        }
#pragma unroll
        for (int ks = 24; ks < 32; ++ks) {      // k in [96,128): action
            v2f b  = *(const v2f*)(wb + ks * 64);
            v2f a0 = *(const v2f*)(ac0 + (ks - 24) * 4);
            v2f a1 = *(const v2f*)(ac1 + (ks - 24) * 4);
            c0 = __builtin_amdgcn_wmma_f32_16x16x4_f32(false, a0, false, b, (short)0, c0, false, false);
            c1 = __builtin_amdgcn_wmma_f32_16x16x4_f32(false, a1, false, b, (short)0, c1, false, false);
        }
        float bias = b1[nt * 16 + mloc];
        int col = nt * 16 + mloc;
#pragma unroll
        for (int v = 0; v < 8; ++v) {
            h1[(roff + v)      * H1_STRIDE + col] = fmaxf(c0[v] + bias, 0.f);
            h1[(16 + roff + v) * H1_STRIDE + col] = fmaxf(c1[v] + bias, 0.f);
        }
    }
    // h1 is wave-private LDS: same-wave DS ops are in order (DScnt handled by compiler)

    // ====================== Layer 2 + fused head ===========================
    // out[m] = sum_n relu(h1 @ W2^T + b2)[m,n] * Wc[n] + bc
    v8f hd0 = {};
    v8f hd1 = {};
    for (int nt = 0; nt < 16; ++nt) {
        v8f c0 = {};
        v8f c1 = {};
        const float* wb2 = pw2 + (nt * 64) * 64 + lane * 2;  // L2-resident, coalesced
        const float* ha0 = h1 + mloc * H1_STRIDE + koff;
        const float* ha1 = h1 + (16 + mloc) * H1_STRIDE + koff;
        if (nt < 15) {                       // stay one ntile (16 KB) ahead in L2->WGP$
            const float* nxt = pw2 + ((nt + 1) * 64) * 64;
#pragma unroll
            for (int p = 0; p < 4; ++p)
                __builtin_prefetch(nxt + (lane + p * 32) * 32, 0, 3);
        }
#pragma unroll 4
        for (int ks = 0; ks < 64; ++ks) {
            v2f b  = *(const v2f*)(wb2 + ks * 64);
            v2f a0 = *(const v2f*)(ha0 + ks * 4);            // conflict-free b64
            v2f a1 = *(const v2f*)(ha1 + ks * 4);
            c0 = __builtin_amdgcn_wmma_f32_16x16x4_f32(false, a0, false, b, (short)0, c0, false, false);
            c1 = __builtin_amdgcn_wmma_f32_16x16x4_f32(false, a1, false, b, (short)0, c1, false, false);
        }
        float bias = b2[nt * 16 + mloc];
        float w    = wc[nt * 16 + mloc];
#pragma unroll
        for (int v = 0; v < 8; ++v) {
            hd0[v] = fmaf(fmaxf(c0[v] + bias, 0.f), w, hd0[v]);
            hd1[v] = fmaf(fmaxf(c1[v] + bias, 0.f), w, hd1[v]);
        }
    }

    // reduce the head dot-product across each 16-lane half (N dimension)
#pragma unroll
    for (int off = 8; off >= 1; off >>= 1) {
#pragma unroll
        for (int v = 0; v < 8; ++v) {
            hd0[v] += __shfl_xor(hd0[v], off, 16);
            hd1[v] += __shfl_xor(hd1[v], off, 16);
        }
    }
    float bc = *bcp;
    if (mloc == 0) {  // lanes 0 and 16 hold the reduced sums for their half
#pragma unroll
        for (int v = 0; v < 8; ++v) {
            out[rowbase + roff + v]      = hd0[v] + bc;   // rows  roff..roff+7  of tile 0
            out[rowbase + 16 + roff + v] = hd1[v] + bc;   // rows 16+roff..      of tile 1
        }
    }
}

// ---------------------------------------------------------------------------
extern "C" void kernel_launch(void* const* d_in, const int* in_sizes, int n_in,
                              void* d_out, int out_size, void* d_ws, size_t ws_size,
                              hipStream_t stream) {
    const float* state  = (const float*)d_in[0];
    const float* action = (const float*)d_in[1];
    const float* alphas = (const float*)d_in[2];
    const float* W1     = (const float*)d_in[3];
    const float* b1     = (const float*)d_in[4];
    const float* W2     = (const float*)d_in[5];
    const float* b2     = (const float*)d_in[6];
    const float* Wh     = (const float*)d_in[7];
    const float* bh     = (const float*)d_in[8];
    float* out = (float*)d_out;

    // workspace: [Wc 256][bc 1][pad][packW1 32768][packW2 65536] (floats)
    float* wsf = (float*)d_ws;
    float* wc  = wsf;
    float* bc  = wsf + 256;
    float* pw1 = wsf + 512;
    float* pw2 = wsf + 512 + PW1_DW;

    critic_prep_head<<<1, 256, 0, stream>>>(alphas, Wh, bh, wc, bc);
    critic_prep_pack<<<(PW1_DW + PW2_DW) / 256, 256, 0, stream>>>(W1, W2, pw1, pw2);

    const size_t smem_bytes = SMEM_DW * sizeof(float);   // 258 KB of the 320 KB WGP LDS
    critic_main<<<BATCH / 128, 128, smem_bytes, stream>>>(
        state, action, b1, b2, pw1, pw2, wc, bc, out);
}